// GraphKoopmanEncoder_57088705298762
// MI455X (gfx1250) — compile-verified
//
#include <hip/hip_runtime.h>
#include <hip/hip_bf16.h>

// ---------------------------------------------------------------------------
// GCN encoder for MI455X (gfx1250, wave32).
// Sizes fixed by the reference: DIN=128, H=64, DEMB=32, G=256.
// N and E are taken from in_sizes.
//
// Roofline: edge aggregation dominates (~3.3 GB of L2-resident traffic,
// ~140 us at HBM peak); GEMMs total ~2.5 GFLOP -> fp32 WMMA
// (v_wmma_f32_16x16x4_f32) keeps exact fp32 semantics at negligible cost.
// ---------------------------------------------------------------------------

typedef float v2f __attribute__((ext_vector_type(2)));
typedef float v8f __attribute__((ext_vector_type(8)));

#define DIN    128
#define HDIM   64
#define DEMB   32
#define NGRAPH 256

// ---------------------------- utility kernels ------------------------------

__global__ void fill_kernel(float* __restrict__ p, float v, long long n) {
    long long t = (long long)blockIdx.x * blockDim.x + threadIdx.x;
    if (t < n) p[t] = v;
}

__global__ void degree_kernel(const int* __restrict__ dst,
                              float* __restrict__ deg, int E) {
    int e = blockIdx.x * blockDim.x + threadIdx.x;
    if (e < E) unsafeAtomicAdd(&deg[dst[e]], 1.0f);
}

__global__ void dinv_kernel(const float* __restrict__ deg,
                            float* __restrict__ dinv, int N) {
    int t = blockIdx.x * blockDim.x + threadIdx.x;
    if (t < N) dinv[t] = rsqrtf(deg[t]);
}

// ------------------------- WMMA fp32 GEMM (16x16 tile / wave) --------------
// C[M x Nc] = A[M x K] * B[K x Nc] (+ bias per column).  M,Nc multiples of
// 16, K multiple of 4.  One wave per 16x16 tile, V_WMMA_F32_16X16X4_F32.
// Lane layout (ISA 7.12.2, 32-bit):
//   A 16x4 : lanes 0-15 hold K=0,1 ; lanes 16-31 hold K=2,3 (M = lane%16)
//   B 4x16 : lanes 0-15 hold K=0,1 ; lanes 16-31 hold K=2,3 (N = lane%16)
//   C 16x16: VGPR v -> M = v + 8*(lane/16), N = lane%16

__global__ void wmma_gemm_f32(const float* __restrict__ A, int lda,
                              const float* __restrict__ B, int ldb,
                              const float* __restrict__ bias,
                              float* __restrict__ C, int ldc,
                              int M, int K, int Nc) {
    const int lane  = threadIdx.x & 31;
    const int wave  = threadIdx.x >> 5;
    const int wpb   = blockDim.x >> 5;
    const int half  = lane >> 4;      // 0: K={0,1}, 1: K={2,3}
    const int lr    = lane & 15;
    const int tilesN = Nc >> 4;
    const int nTiles = (M >> 4) * tilesN;

    for (int tile = blockIdx.x * wpb + wave; tile < nTiles;
         tile += gridDim.x * wpb) {
        const int tm  = tile / tilesN;
        const int tn  = tile - tm * tilesN;
        const int row = (tm << 4) + lr;   // A row for this lane
        const int col = (tn << 4) + lr;   // B/C column for this lane

        v8f acc = {};
        const float* pa = A + (size_t)row * lda + 2 * half;   // step +4
        const float* pb = B + (size_t)(2 * half) * ldb + col; // step +4*ldb
        const int ldb4 = 4 * ldb;

        for (int k = 0; k < K; k += 4) {
            v2f a = *(const v2f*)pa;      // K, K+1 for this half-wave
            v2f b;
            b.x = pb[0];
            b.y = pb[ldb];
            acc = __builtin_amdgcn_wmma_f32_16x16x4_f32(
                /*neg_a=*/false, a, /*neg_b=*/false, b,
                /*c_mod=*/(short)0, acc, /*reuse_a=*/false, /*reuse_b=*/false);
            pa += 4;
            pb += ldb4;
        }

        float bv = bias ? bias[col] : 0.0f;
#pragma unroll
        for (int v = 0; v < 8; ++v) {
            int orow = (tm << 4) + v + 8 * half;
            C[(size_t)orow * ldc + col] = acc[v] + bv;
        }
    }
}

// -------------------------- edge aggregation -------------------------------
// One wave (32 lanes) per edge; each lane owns 2 consecutive features.
// Edge endpoints made wave-uniform via readfirstlane so dinv/coef and the
// row bases use the scalar path (saddr-form global ops), then:
//   per lane: one global_load_b64 of h[src], two global_atomic_add_f32.

__global__ void edge_aggregate(const int* __restrict__ src,
                               const int* __restrict__ dst,
                               const float* __restrict__ h,
                               const float* __restrict__ dinv,
                               float* __restrict__ agg, int E) {
    const int lane = threadIdx.x & 31;
    const int wave = threadIdx.x >> 5;
    const int wpb  = blockDim.x >> 5;
    int e = blockIdx.x * wpb + wave;
    if (e >= E) return;

    int s = __builtin_amdgcn_readfirstlane(src[e]);
    int d = __builtin_amdgcn_readfirstlane(dst[e]);
    float coef = dinv[s] * dinv[d];                 // wave-uniform

    const v2f* hs = (const v2f*)(h + (size_t)s * HDIM);
    v2f v = hs[lane];
    float* ad = agg + (size_t)d * HDIM + 2 * lane;
    unsafeAtomicAdd(ad,     v.x * coef);
    unsafeAtomicAdd(ad + 1, v.y * coef);
}

// self-loop message + bias + ReLU, in place on agg
__global__ void post_agg_relu(float* __restrict__ agg,
                              const float* __restrict__ h,
                              const float* __restrict__ dinv,
                              const float* __restrict__ bias, int N) {
    long long t = (long long)blockIdx.x * blockDim.x + threadIdx.x;
    if (t >= (long long)N * HDIM) return;
    int n = (int)(t >> 6);
    int f = (int)(t & 63);
    float di = dinv[n];
    float v = agg[t] + h[t] * di * di + bias[f];
    agg[t] = fmaxf(v, 0.0f);
}

// ------------------------------ pooling ------------------------------------
// One wave per node; scalar batch id; 2 features per lane.

__global__ void pool_kernel(const float* __restrict__ h,
                            const int* __restrict__ batch,
                            float* __restrict__ sums,
                            float* __restrict__ cnt, int N) {
    const int lane = threadIdx.x & 31;
    const int wave = threadIdx.x >> 5;
    const int wpb  = blockDim.x >> 5;
    int n = blockIdx.x * wpb + wave;
    if (n >= N) return;

    int g = __builtin_amdgcn_readfirstlane(batch[n]);
    const v2f* hn = (const v2f*)(h + (size_t)n * HDIM);
    v2f v = hn[lane];
    float* sg = sums + (size_t)g * HDIM + 2 * lane;
    unsafeAtomicAdd(sg,     v.x);
    unsafeAtomicAdd(sg + 1, v.y);
    if (lane == 0) unsafeAtomicAdd(&cnt[g], 1.0f);
}

__global__ void pool_divide(float* __restrict__ sums,
                            const float* __restrict__ cnt, int G) {
    int t = blockIdx.x * blockDim.x + threadIdx.x;
    if (t >= G * HDIM) return;
    int g = t >> 6;
    sums[t] /= fmaxf(cnt[g], 1.0f);
}

// ------------------------------ launch -------------------------------------

extern "C" void kernel_launch(void* const* d_in, const int* in_sizes, int n_in,
                              void* d_out, int out_size, void* d_ws, size_t ws_size,
                              hipStream_t stream) {
    const float* x     = (const float*)d_in[0];          // [N, 128]
    const int*   eidx  = (const int*)  d_in[1];          // [2, E]
    const int*   batch = (const int*)  d_in[2];          // [N]
    // d_in[3] = num_graphs (fixed 256 in reference)
    const float* W1 = (const float*)d_in[4];             // [128, 64]
    const float* b1 = (const float*)d_in[5];             // [64]
    const float* W2 = (const float*)d_in[6];             // [64, 64]
    const float* b2 = (const float*)d_in[7];             // [64]
    const float* Wl = (const float*)d_in[8];             // [64, 32]
    const float* bl = (const float*)d_in[9];             // [32]
    float* out = (float*)d_out;                          // [256, 32]

    const int N = in_sizes[0] / DIN;
    const int E = in_sizes[1] / 2;
    const int G = NGRAPH;
    const int* src = eidx;
    const int* dst = eidx + E;

    // ---- workspace layout (floats) ----
    float* ws   = (float*)d_ws;
    float* deg  = ws;                       // N
    float* dinv = deg  + N;                 // N
    float* h1   = dinv + N;                 // N*64  (GEMM outputs, reused)
    float* agg1 = h1   + (size_t)N * HDIM;  // N*64
    float* agg2 = agg1 + (size_t)N * HDIM;  // N*64
    float* pool = agg2 + (size_t)N * HDIM;  // G*64
    float* cnt  = pool + (size_t)G * HDIM;  // G
    const long long zeroCount = 2LL * N * HDIM + G * HDIM + G; // agg1..cnt contiguous

    const int BLK = 256;
    const int WPB = BLK / 32;               // waves per block
    auto blocks = [](long long work, int per) { return (int)((work + per - 1) / per); };

    // 1) deg = 1 (self loop), zero accumulators (fresh every call -> replay-safe)
    fill_kernel<<<blocks(N, BLK), BLK, 0, stream>>>(deg, 1.0f, N);
    fill_kernel<<<blocks(zeroCount, BLK), BLK, 0, stream>>>(agg1, 0.0f, zeroCount);

    // 2) degrees + dinv
    degree_kernel<<<blocks(E, BLK), BLK, 0, stream>>>(dst, deg, E);
    dinv_kernel<<<blocks(N, BLK), BLK, 0, stream>>>(deg, dinv, N);

    // 3) layer 1 GEMM: h1 = x @ W1   (M=N, K=128, Nc=64)
    {
        int tiles = (N >> 4) * (HDIM >> 4);
        wmma_gemm_f32<<<blocks(tiles, WPB), BLK, 0, stream>>>(
            x, DIN, W1, HDIM, nullptr, h1, HDIM, N, DIN, HDIM);
    }

    // 4) edge aggregation layer 1 (one wave per edge)
    edge_aggregate<<<blocks(E, WPB), BLK, 0, stream>>>(src, dst, h1, dinv, agg1, E);

    // 5) self-loop + bias + ReLU  -> agg1 holds layer-1 activations
    post_agg_relu<<<blocks((long long)N * HDIM, BLK), BLK, 0, stream>>>(
        agg1, h1, dinv, b1, N);

    // 6) layer 2 GEMM: h1 = agg1 @ W2   (M=N, K=64, Nc=64)
    {
        int tiles = (N >> 4) * (HDIM >> 4);
        wmma_gemm_f32<<<blocks(tiles, WPB), BLK, 0, stream>>>(
            agg1, HDIM, W2, HDIM, nullptr, h1, HDIM, N, HDIM, HDIM);
    }

    // 7) edge aggregation layer 2
    edge_aggregate<<<blocks(E, WPB), BLK, 0, stream>>>(src, dst, h1, dinv, agg2, E);

    // 8) self-loop + bias + ReLU -> agg2 holds layer-2 activations
    post_agg_relu<<<blocks((long long)N * HDIM, BLK), BLK, 0, stream>>>(
        agg2, h1, dinv, b2, N);

    // 9) global mean pool (one wave per node)
    pool_kernel<<<blocks(N, WPB), BLK, 0, stream>>>(agg2, batch, pool, cnt, N);
    pool_divide<<<blocks(G * HDIM, BLK), BLK, 0, stream>>>(pool, cnt, G);

    // 10) final linear: out = pool @ Wl + bl   (M=256, K=64, Nc=32)
    {
        int tiles = (G >> 4) * (DEMB >> 4);   // 32 tiles
        wmma_gemm_f32<<<blocks(tiles, WPB), BLK, 0, stream>>>(
            pool, HDIM, Wl, DEMB, bl, out, DEMB, G, HDIM, DEMB);
    }
}